// EdgeCGBlock_13391708029791
// MI455X (gfx1250) — compile-verified
//
#include <hip/hip_runtime.h>

// ============================================================================
// EdgeCGBlock (e3nn tensor product message passing) for MI455X / gfx1250
//
// Heavy math:  out[e,w,k] = sum_i (node[e,:,i] @ (alpha*W)) * wgt[e,v] * (Y.C3J)
// The u-contraction is an edge-independent-weight GEMM -> V_WMMA_F32_16X16X4_F32
// (fp32 matches reference precision; 16-edge M tiles on wave32).
// ============================================================================

typedef float v2f __attribute__((ext_vector_type(2)));
typedef float v8f __attribute__((ext_vector_type(8)));

// ---------------- path metadata (PATHS order of the reference) -------------
// PATHS = [(0,0,0),(1,1,0),(2,2,0),(0,1,1),(1,0,1),(1,2,1),(2,1,1),
//          (0,2,2),(2,0,2),(1,1,2),(2,2,2)]
__constant__ int PATH_L1_C[11]   = {0,1,2,0,1,1,2,0,2,1,2};
__constant__ int PATH_L2_C[11]   = {0,1,2,1,0,2,1,2,0,1,2};
__constant__ int PATH_L3_C[11]   = {0,0,0,1,1,1,1,2,2,2,2};
// D-vector offsets: sizes (2l1+1)(2l3+1) = {1,3,5,3,9,9,15,5,25,15,25}
__constant__ int PATH_DOFF_C[12] = {0,1,4,9,12,21,30,45,50,75,90,115};
// tp_w flat offsets (cumsum of u*4*w per path)
__constant__ int PATH_WOFF_C[11] = {0,16384,24576,28672,36864,40960,45056,
                                    47104,51200,52224,54272};
// alpha = 1/sqrt(FAN_IN[l3]) ; FAN_IN = {448,576,512}
__constant__ float ALPHA_C[3] = {0.047245559f, 0.041666667f, 0.044194174f};

// Wpack geometry (floats): l1 groups, Ntiles x Kchunks x 64
//   l1=0: 28 tiles * 16 kc   (u=64, Ncols=448)  -> 28672
//   l1=1: 36 tiles *  8 kc   (u=32, Ncols=576)  -> 18432
//   l1=2: 32 tiles *  4 kc   (u=16, Ncols=512)  ->  8192
#define WP_BASE0 0
#define WP_BASE1 28672
#define WP_BASE2 47104
#define WP_TOTAL 55296

// N-tile (16 columns) -> (path, v, w_base); every 16-col tile lies in one (p,v)
__device__ __forceinline__ void tileMeta(int l1, int nt, int& p, int& v, int& wb) {
  if (l1 == 0) {
    if      (nt < 16) { p = 0; v = nt >> 2;            wb = (nt & 3) << 4; }
    else if (nt < 24) { int t = nt - 16; p = 3; v = t >> 1; wb = (t & 1) << 4; }
    else              { p = 7; v = nt - 24;            wb = 0; }
  } else if (l1 == 1) {
    if      (nt < 16) { p = 1; v = nt >> 2;            wb = (nt & 3) << 4; }
    else if (nt < 24) { int t = nt - 16; p = 4; v = t >> 1; wb = (t & 1) << 4; }
    else if (nt < 32) { int t = nt - 24; p = 5; v = t >> 1; wb = (t & 1) << 4; }
    else              { p = 9; v = nt - 32;            wb = 0; }
  } else {
    if      (nt < 16) { p = 2; v = nt >> 2;            wb = (nt & 3) << 4; }
    else if (nt < 24) { int t = nt - 16; p = 6; v = t >> 1; wb = (t & 1) << 4; }
    else if (nt < 28) { p = 8; v = nt - 24;            wb = 0; }
    else              { p = 10; v = nt - 28;           wb = 0; }
  }
}

// ============================================================================
// Kernel 0: real Wigner-3j tables (fp64, runs once, 11 threads)
// ============================================================================
struct cplx { double re, im; };
__device__ __forceinline__ cplx cmul(cplx a, cplx b) {
  return {a.re * b.re - a.im * b.im, a.re * b.im + a.im * b.re};
}
__device__ double dfact(int n) { double r = 1.0; for (int i = 2; i <= n; ++i) r *= (double)i; return r; }

__device__ double su2_cg(int j1, int m1, int j2, int m2, int j3, int m3) {
  if (m3 != m1 + m2) return 0.0;
  int vmin = 0;
  if (-j1 + j2 + m3 > vmin) vmin = -j1 + j2 + m3;
  if (-j1 + m1 > vmin)      vmin = -j1 + m1;
  int vmax = j2 + j3 + m1;
  if (j3 - j1 + j2 < vmax) vmax = j3 - j1 + j2;
  if (j3 + m3 < vmax)      vmax = j3 + m3;
  double Cn = sqrt((2.0 * j3 + 1.0) * dfact(j3 + j1 - j2) * dfact(j3 - j1 + j2) *
                   dfact(j1 + j2 - j3) * dfact(j3 + m3) * dfact(j3 - m3) /
                   (dfact(j1 + j2 + j3 + 1) * dfact(j1 - m1) * dfact(j1 + m1) *
                    dfact(j2 - m2) * dfact(j2 + m2)));
  double S = 0.0;
  for (int v = vmin; v <= vmax; ++v) {
    double sg = ((v + j2 + m2) & 1) ? -1.0 : 1.0;
    S += sg * dfact(j2 + j3 + m1 - v) * dfact(j1 - m1 + v) /
         (dfact(v) * dfact(j3 - j1 + j2 - v) * dfact(j3 + m3 - v) * dfact(v + j1 - j2 - m3));
  }
  return Cn * S;
}

__device__ void r2c(int l, cplx q[5][5]) {
  for (int a = 0; a < 5; ++a) for (int b = 0; b < 5; ++b) q[a][b] = {0.0, 0.0};
  const double is2 = 0.7071067811865476;
  for (int m = -l; m < 0; ++m) { q[l + m][l - m].re = is2; q[l + m][l + m].im = -is2; }
  q[l][l].re = 1.0;
  for (int m = 1; m <= l; ++m) {
    double sg = (m & 1) ? -1.0 : 1.0;
    q[l + m][l + m].re = sg * is2;
    q[l + m][l - m].im = sg * is2;
  }
  if (l == 1) {        // * (-i)
    for (int a = 0; a < 5; ++a) for (int b = 0; b < 5; ++b) {
      double re = q[a][b].re, im = q[a][b].im;
      q[a][b].re = im; q[a][b].im = -re;
    }
  } else if (l == 2) { // * (-1)
    for (int a = 0; a < 5; ++a) for (int b = 0; b < 5; ++b) {
      q[a][b].re = -q[a][b].re; q[a][b].im = -q[a][b].im;
    }
  }
}

__global__ void w3j_kernel(float* __restrict__ out) {
  int p = threadIdx.x;
  if (p >= 11) return;
  const int l1 = PATH_L1_C[p], l2 = PATH_L2_C[p], l3 = PATH_L3_C[p];
  const int n1 = 2 * l1 + 1, n2 = 2 * l2 + 1, n3 = 2 * l3 + 1;
  double C[5][5][5];
  for (int a = 0; a < n1; ++a)
    for (int b = 0; b < n2; ++b)
      for (int c = 0; c < n3; ++c)
        C[a][b][c] = su2_cg(l1, a - l1, l2, b - l2, l3, c - l3);
  cplx Q1[5][5], Q2[5][5], Q3[5][5];
  r2c(l1, Q1); r2c(l2, Q2); r2c(l3, Q3);
  double Cr[5][5][5], Ci[5][5][5], nr = 0.0, ni = 0.0;
  for (int jj = 0; jj < n1; ++jj)
    for (int ll = 0; ll < n2; ++ll)
      for (int mm = 0; mm < n3; ++mm) {
        cplx s = {0.0, 0.0};
        for (int i = 0; i < n1; ++i)
          for (int k = 0; k < n2; ++k)
            for (int n = 0; n < n3; ++n) {
              cplx t = cmul(Q1[i][jj], Q2[k][ll]);
              cplx q3c = {Q3[n][mm].re, -Q3[n][mm].im};
              t = cmul(t, q3c);
              s.re += t.re * C[i][k][n];
              s.im += t.im * C[i][k][n];
            }
        Cr[jj][ll][mm] = s.re; Ci[jj][ll][mm] = s.im;
        nr += s.re * s.re; ni += s.im * s.im;
      }
  bool useIm = (sqrt(nr) < 1e-8);
  double nrm = useIm ? sqrt(ni) : sqrt(nr);
  if (nrm < 1e-30) nrm = 1.0;
  for (int i = 0; i < 5; ++i)
    for (int j = 0; j < 5; ++j)
      for (int k = 0; k < 5; ++k) {
        float v = 0.f;
        if (i < n1 && j < n2 && k < n3)
          v = (float)((useIm ? Ci[i][j][k] : Cr[i][j][k]) / nrm);
        out[p * 125 + (i * 5 + j) * 5 + k] = v;
      }
}

// ============================================================================
// Kernel W: repack tp_w into WMMA B-fragment order, alpha folded in.
// Layout: Wpack[l1base + ((nt*KC + kc)*2 + vgpr)*32 + lane]
//   element = alpha * W[k][col],  k = kc*4 + vgpr + 2*(lane>=16), col = nt*16 + lane%16
// ============================================================================
__global__ void repack_kernel(const float* __restrict__ tp_w, float* __restrict__ Wpack) {
  int idx = blockIdx.x * blockDim.x + threadIdx.x;
  if (idx >= WP_TOTAL) return;
  int l1, local;
  if      (idx < WP_BASE1) { l1 = 0; local = idx; }
  else if (idx < WP_BASE2) { l1 = 1; local = idx - WP_BASE1; }
  else                     { l1 = 2; local = idx - WP_BASE2; }
  const int KC = (l1 == 0) ? 16 : (l1 == 1) ? 8 : 4;
  int nt  = local / (KC * 64);
  int rem = local % (KC * 64);
  int kc = rem >> 6;
  int r  = (rem >> 5) & 1;
  int ln = rem & 31;
  int n = ln & 15, hf = ln >> 4;
  int k = kc * 4 + r + 2 * hf;
  int p, v, wb;
  tileMeta(l1, nt, p, v, wb);
  int l3 = PATH_L3_C[p];
  int wp = 64 >> l3;                 // MSG_MULS = {64,32,16}
  int w = wb + n;
  Wpack[idx] = tp_w[PATH_WOFF_C[p] + (k * 4 + v) * wp + w] * ALPHA_C[l3];
}

// ============================================================================
// Kernel 1: per-edge RBF + radial MLPs -> wgt[e][l][v] (mask folded) and Y[e][9]
// ============================================================================
__global__ void edge_kernel(const float* __restrict__ edge_vec,
                            const float* __restrict__ mask,
                            const float* __restrict__ w1, const float* __restrict__ b1,
                            const float* __restrict__ w2, const float* __restrict__ b2,
                            float* __restrict__ wgt_g, float* __restrict__ Y_g, int E) {
  __shared__ float sw1[3 * 32 * 32];
  __shared__ float sb1[3 * 32];
  __shared__ float sw2[3 * 32 * 4];
  __shared__ float sb2[3 * 4];
  const int tid = threadIdx.x;
  for (int i = tid; i < 3072; i += blockDim.x) sw1[i] = w1[i];
  for (int i = tid; i < 96;   i += blockDim.x) sb1[i] = b1[i];
  for (int i = tid; i < 384;  i += blockDim.x) sw2[i] = w2[i];
  for (int i = tid; i < 12;   i += blockDim.x) sb2[i] = b2[i];
  __syncthreads();
  int e = blockIdx.x * blockDim.x + tid;
  if (e >= E) return;

  float vx = edge_vec[(size_t)e * 3 + 0];
  float vy = edge_vec[(size_t)e * 3 + 1];
  float vz = edge_vec[(size_t)e * 3 + 2];
  float r = sqrtf(vx * vx + vy * vy + vz * vz);
  r = fmaxf(r, 1e-12f);
  float xh = vx / r, yh = vy / r, zh = vz / r;

  const float PI_F = 3.14159265358979f;
  float rb  = 0.5f * (cosf(r * (PI_F / 5.0f)) + 1.0f);
  float cut = (r < 5.0f) ? rb * rb : 0.0f;   // rbounds applied twice, + cutoff
  float er  = expf(-r);
  const float E5   = 0.006737946999085467f;  // exp(-5)
  const float STEP = (1.0f - E5) / 31.0f;
  const float BW   = 0.0625f * (1.0f - E5);  // 2/32*(1-exp(-5))
  const float BETA = 1.0f / (BW * BW);
  float rbf[32];
#pragma unroll
  for (int i = 0; i < 32; ++i) {
    float d = er - (E5 + (float)i * STEP);
    rbf[i] = cut * expf(-BETA * d * d);
  }

  float msk = mask[e];
  for (int l = 0; l < 3; ++l) {
    float h[32];
    for (int o = 0; o < 32; ++o) {
      float acc = sb1[l * 32 + o];
      const float* wrow = &sw1[l * 1024 + o];
#pragma unroll 8
      for (int j = 0; j < 32; ++j) acc += rbf[j] * wrow[j * 32];
      h[o] = acc / (1.0f + expf(-acc));   // silu
    }
    for (int v = 0; v < 4; ++v) {
      float acc = sb2[l * 4 + v];
#pragma unroll 8
      for (int o = 0; o < 32; ++o) acc += h[o] * sw2[l * 128 + o * 4 + v];
      wgt_g[(size_t)e * 12 + l * 4 + v] = acc * msk;   // fold mask into v-weights
    }
  }

  const float s3 = 1.7320508075688772f, s5 = 2.2360679774997896f, s15 = 3.872983346207417f;
  float* Y = Y_g + (size_t)e * 9;
  Y[0] = 1.0f;
  Y[1] = s3 * xh; Y[2] = s3 * yh; Y[3] = s3 * zh;
  Y[4] = s15 * xh * zh;
  Y[5] = s15 * xh * yh;
  Y[6] = s5 * (yh * yh - 0.5f * (xh * xh + zh * zh));
  Y[7] = s15 * yh * zh;
  Y[8] = 0.5f * s15 * (zh * zh - xh * xh);
}

// ============================================================================
// Kernel 2: the tensor product. 1 wave = 16 edges (WMMA M tile).
// ============================================================================
template <int L1>
__device__ void tp_l1(const float* __restrict__ node_feat, const int* smSrc,
                      const float* smW, const float* smWgt, const float* smD,
                      float* smOut, int lane) {
  constexpr int U    = (L1 == 0) ? 64 : (L1 == 1) ? 32 : 16;
  constexpr int NI   = 2 * L1 + 1;
  constexpr int KC   = U / 4;
  constexpr int NT   = (L1 == 0) ? 28 : (L1 == 1) ? 36 : 32;
  constexpr int NOFF = (L1 == 0) ? 0 : (L1 == 1) ? 64 : 160;
  const int m = lane & 15, hf = lane >> 4;
  const float* row = node_feat + (size_t)smSrc[m] * 240 + NOFF;

  // A panel: A[m][k] = node[e_m, u=k, i]; lane layout per V_WMMA_F32_16X16X4_F32
  float a0[NI][KC], a1[NI][KC];
#pragma unroll
  for (int i = 0; i < NI; ++i)
#pragma unroll
    for (int kc = 0; kc < KC; ++kc) {
      int k0 = kc * 4 + 2 * hf;
      a0[i][kc] = row[(k0 + 0) * NI + i];
      a1[i][kc] = row[(k0 + 1) * NI + i];
    }

  for (int nt = 0; nt < NT; ++nt) {
    v8f acc[NI];
#pragma unroll
    for (int i = 0; i < NI; ++i) acc[i] = (v8f){0.f, 0.f, 0.f, 0.f, 0.f, 0.f, 0.f, 0.f};
    const float* wb = smW + (size_t)nt * KC * 64;
#pragma unroll
    for (int kc = 0; kc < KC; ++kc) {
      v2f B = {wb[kc * 64 + lane], wb[kc * 64 + 32 + lane]};  // B fragment reused over i
#pragma unroll
      for (int i = 0; i < NI; ++i) {
        v2f A = {a0[i][kc], a1[i][kc]};
        acc[i] = __builtin_amdgcn_wmma_f32_16x16x4_f32(false, A, false, B,
                                                       (short)0, acc[i], false, false);
      }
    }
    // ---- stage 2: contract with wgt[e,v] and D[e, path, i, k] -> out tile ----
    int p, v, wbs;
    tileMeta(L1, nt, p, v, wbs);
    int l2 = PATH_L2_C[p], l3 = PATH_L3_C[p];
    int K3 = 2 * l3 + 1;
    int outbase = ((l3 == 0) ? 0 : (l3 == 1) ? 64 : 160) + (wbs + m) * K3;  // m = lane&15 = col n
    int doffp = PATH_DOFF_C[p];
#pragma unroll
    for (int i = 0; i < NI; ++i) {
#pragma unroll
      for (int r = 0; r < 8; ++r) {
        int mm = r + 8 * hf;  // C layout: vgpr r <-> row r (lanes<16) / r+8 (lanes>=16)
        float s = smWgt[mm * 12 + l2 * 4 + v] * acc[i][r];
        const float* dp = smD + mm * 115 + doffp + i * K3;
        float* op = smOut + mm * 240 + outbase;
        for (int k = 0; k < K3; ++k) op[k] += s * dp[k];  // each (mm,w,k) owned by one lane
      }
    }
  }
}

__global__ void __launch_bounds__(128)
tp_kernel(const float* __restrict__ node_feat, const int* __restrict__ edge_src,
          const float* __restrict__ wsW3J, const float* __restrict__ Wpack,
          const float* __restrict__ wgt_g, const float* __restrict__ Y_g,
          float* __restrict__ out, int E) {
  extern __shared__ float sm[];
  float* smW   = sm;          // weight panel for current l1 (max 28672 floats)
  float* smW3J = sm + 28672;  // 1375 floats
  const int tid = threadIdx.x, lane = tid & 31, wv = tid >> 5;
  float* base  = sm + 30048 + wv * 6032;   // per-wave scratch
  int*   smSrc = (int*)base;               // 16
  float* smWgt = base + 16;                // 192
  float* smY   = base + 208;               // 144
  float* smD   = base + 352;               // 1840
  float* smOut = base + 2192;              // 3840

  int ntiles = (E + 15) >> 4;
  int tile = blockIdx.x * 4 + wv;
  if (tile >= ntiles) tile = ntiles - 1;          // benign duplicate on ragged grids
  size_t e0 = (size_t)tile * 16;
  if (e0 + 16 > (size_t)E) e0 = (size_t)E - 16;

  for (int i = tid; i < 1375; i += blockDim.x) smW3J[i] = wsW3J[i];
  if (lane < 16) smSrc[lane] = edge_src[e0 + lane];
  for (int i = lane; i < 192;  i += 32) smWgt[i] = wgt_g[e0 * 12 + i];
  for (int i = lane; i < 144;  i += 32) smY[i]   = Y_g[e0 * 9 + i];
  for (int i = lane; i < 3840; i += 32) smOut[i] = 0.f;
  __syncthreads();

  // D[e, path, i, k] = sum_j Y_l2[e,j] * C3J_path[i,j,k]   (alpha already in Wpack)
  for (int t = lane; t < 16 * 115; t += 32) {
    int m = t / 115, u = t - m * 115;
    int p = 0;
    while (p < 10 && PATH_DOFF_C[p + 1] <= u) ++p;
    int local = u - PATH_DOFF_C[p];
    int l3 = PATH_L3_C[p]; int K3 = 2 * l3 + 1;
    int i2 = local / K3, kk = local - i2 * K3;
    int l2 = PATH_L2_C[p]; int NJ = 2 * l2 + 1; int yo = l2 * l2;
    float s = 0.f;
    for (int j = 0; j < NJ; ++j)
      s += smY[m * 9 + yo + j] * smW3J[p * 125 + (i2 * 5 + j) * 5 + kk];
    smD[t] = s;
  }

  // ---- l1 = 0 ----
  __syncthreads();
  for (int i = tid; i < 28672; i += blockDim.x) smW[i] = Wpack[WP_BASE0 + i];
  __syncthreads();
  tp_l1<0>(node_feat, smSrc, smW, smWgt, smD, smOut, lane);
  // ---- l1 = 1 ----
  __syncthreads();
  for (int i = tid; i < 18432; i += blockDim.x) smW[i] = Wpack[WP_BASE1 + i];
  __syncthreads();
  tp_l1<1>(node_feat, smSrc, smW, smWgt, smD, smOut, lane);
  // ---- l1 = 2 ----
  __syncthreads();
  for (int i = tid; i < 8192; i += blockDim.x) smW[i] = Wpack[WP_BASE2 + i];
  __syncthreads();
  tp_l1<2>(node_feat, smSrc, smW, smWgt, smD, smOut, lane);

  for (int i = lane; i < 3840; i += 32) out[e0 * 240 + i] = smOut[i];
}

// ============================================================================
// Host launcher
// ============================================================================
extern "C" void kernel_launch(void* const* d_in, const int* in_sizes, int n_in,
                              void* d_out, int out_size, void* d_ws, size_t ws_size,
                              hipStream_t stream) {
  const float* node_feat  = (const float*)d_in[0];
  const int*   edge_index = (const int*)d_in[1];   // (2,E) row-major; row 0 = src
  const float* edge_vec   = (const float*)d_in[2];
  const float* mask       = (const float*)d_in[3];
  const float* w1         = (const float*)d_in[4];
  const float* b1         = (const float*)d_in[5];
  const float* w2         = (const float*)d_in[6];
  const float* b2         = (const float*)d_in[7];
  const float* tp_w       = (const float*)d_in[8];
  const int E = in_sizes[3];                       // mask element count

  // workspace layout (floats): [0,2048) W3J | [2048,57344) Wpack |
  //                            [57344, +12E) wgt | then 9E Y   (~68 MB total)
  float* ws     = (float*)d_ws;
  float* wsW3J  = ws;
  float* wsWp   = ws + 2048;
  float* wsWgt  = ws + 57344;
  float* wsY    = wsWgt + (size_t)E * 12;

  w3j_kernel<<<1, 32, 0, stream>>>(wsW3J);
  repack_kernel<<<(WP_TOTAL + 255) / 256, 256, 0, stream>>>(tp_w, wsWp);
  edge_kernel<<<(E + 255) / 256, 256, 0, stream>>>(edge_vec, mask, w1, b1, w2, b2,
                                                   wsWgt, wsY, E);
  int ntiles  = (E + 15) / 16;
  int nblocks = (ntiles + 3) / 4;
  size_t shbytes = (size_t)(30048 + 4 * 6032) * sizeof(float);  // 216704 B (< 320 KB WGP LDS)
  tp_kernel<<<nblocks, 128, shbytes, stream>>>(node_feat, edge_index, wsW3J, wsWp,
                                               wsWgt, wsY, (float*)d_out, E);
}